// EmbeddingModel_89034672046590
// MI455X (gfx1250) — compile-verified
//
#include <hip/hip_runtime.h>
#include <stdint.h>

#define N_ROWS 4096
#define V_DIM  10000
#define D_DIM  128
#define MROWS  32                         // rows per block
#define KC     16
#define KST    20                         // padded K stride (floats): bank-conflict-free, rows stay 16B-aligned
#define NCH    (V_DIM / KC)               // 625 chunks
#define A_FLOATS   (3 * MROWS * KST)      // 1920
#define B_FLOATS   (2 * D_DIM * KST)      // 5120
#define BUF_FLOATS (A_FLOATS + B_FLOATS)  // 7040 floats = 28160 B per buffer
#define REAL_ITEMS 1408                   // (3*32 + 2*128) * KC/4 float4 items per chunk
#define ITEMS_PT   6                      // padded to 1536 = 6 * 256
#define MARGIN 0.5f

typedef __attribute__((ext_vector_type(2))) float v2f;
typedef __attribute__((ext_vector_type(8))) float v8f;

// One block = 8 waves = 256 threads: 32 rows x 128 dims, three GEMMs at once.
// Each wave owns one 16-col tile and TWO 16-row tiles (B fragments reused).
// Async-to-LDS triple-buffered pipeline, depth 2 chunks.
__global__ __launch_bounds__(256) void embed_gemm_kernel(
    const float* __restrict__ ctx,   // [N, V]
    const float* __restrict__ rsp,   // [N, V]
    const float* __restrict__ ngr,   // [N, V]
    const float* __restrict__ embC,  // [D, V]
    const float* __restrict__ embR,  // [D, V]
    float* __restrict__ out)         // [2*N + 1]
{
    __shared__ __attribute__((aligned(16))) float ldsBuf[3][BUF_FLOATS];
    __shared__ float ldsPos[MROWS];
    __shared__ float ldsNeg[MROWS];

    const int tid     = threadIdx.x;
    const int wave    = tid >> 5;
    const int lane    = tid & 31;
    const int l16     = lane & 15;
    const int halfId  = lane >> 4;
    const int rowBase = blockIdx.x * MROWS;
    const int dBase   = wave * 16;

    if (tid < MROWS) { ldsPos[tid] = 0.f; ldsNeg[tid] = 0.f; }

    // ---- precompute the 6 per-thread copy items (float4 granularity) ----
    // Items 0..383:    A tiles (3 matrices x 32 rows x 4 float4 of K).
    // Items 384..1407: B tiles (2 embeddings x 128 d x 4 float4 of K).
    // Items 1408..1535 duplicate items 0..127 so every wave issues exactly 6 asyncs/chunk.
    uint64_t itemG[ITEMS_PT];
    uint32_t itemL[ITEMS_PT];
    #pragma unroll
    for (int j = 0; j < ITEMS_PT; ++j) {
        const int q  = tid + (j << 8);
        const int qa = (q < REAL_ITEMS) ? q : (q - REAL_ITEMS);
        if (qa < 384) {
            const int m  = qa >> 7;
            const int rm = qa & 127;
            const int r  = rm >> 2;
            const int c4 = (rm & 3) << 2;
            const float* src = (m == 0) ? ctx : ((m == 1) ? rsp : ngr);
            itemG[j] = (uint64_t)(uintptr_t)(src + (size_t)(rowBase + r) * V_DIM + c4);
            itemL[j] = (uint32_t)((m * MROWS * KST + r * KST + c4) * 4);
        } else {
            const int qb = qa - 384;
            const int e  = qb >> 9;
            const int rm = qb & 511;
            const int d  = rm >> 2;
            const int c4 = (rm & 3) << 2;
            const float* src = (e == 0) ? embC : embR;
            itemG[j] = (uint64_t)(uintptr_t)(src + (size_t)d * V_DIM + c4);
            itemL[j] = (uint32_t)((A_FLOATS + e * D_DIM * KST + d * KST + c4) * 4);
        }
    }
    const uint32_t ldsBase = (uint32_t)(uintptr_t)&ldsBuf[0][0];

    auto issue_chunk = [&](int ci, int b) {
        const uint32_t bofs = ldsBase + (uint32_t)b * (BUF_FLOATS * 4);
        const uint64_t gofs = (uint64_t)ci * (KC * 4);
        #pragma unroll
        for (int j = 0; j < ITEMS_PT; ++j) {
            const uint32_t l = itemL[j] + bofs;
            const uint64_t g = itemG[j] + gofs;
            asm volatile("global_load_async_to_lds_b128 %0, %1, off"
                         :: "v"(l), "v"(g) : "memory");
        }
    };

    v8f accC[2] = {{}, {}}, accR[2] = {{}, {}}, accN[2] = {{}, {}};

    // ---- pipeline prologue: chunks 0 and 1 in flight ----
    issue_chunk(0, 0);
    issue_chunk(1, 1);

    int bn = 2;  // buffer for next issued chunk (i+2)
    int bc = 0;  // buffer for chunk being computed (i)

    for (int i = 0; i < NCH; ++i) {
        // Retire chunk i (per-wave 6 asyncs of chunk i+1 may stay in flight).
        if (i < NCH - 1) asm volatile("s_wait_asynccnt 6" ::: "memory");
        else             asm volatile("s_wait_asynccnt 0" ::: "memory");
        __syncthreads();   // publish chunk i across waves; also fences last use of buffer bn

        if (i + 2 < NCH) {
            issue_chunk(i + 2, bn);
            bn = (bn == 2) ? 0 : bn + 1;
        }

        const float* bp = &ldsBuf[bc][0];
        bc = (bc == 2) ? 0 : bc + 1;
        const float* aBase = bp;              // A(m,r,k) = m*32*KST + r*KST + k
        const float* bBase = bp + A_FLOATS;   // B(e,d,k) = e*128*KST + d*KST + k

        #pragma unroll
        for (int ks = 0; ks < KC; ks += 4) {
            const int kk = ks + (halfId << 1);     // lane-half picks K pair {0,1} vs {2,3}
            const v2f bCf = *(const v2f*)(bBase + (dBase + l16) * KST + kk);
            const v2f bRf = *(const v2f*)(bBase + D_DIM * KST + (dBase + l16) * KST + kk);
            #pragma unroll
            for (int t = 0; t < 2; ++t) {
                const float* aT = aBase + (t * 16 + l16) * KST + kk;
                const v2f aC = *(const v2f*)(aT + 0 * MROWS * KST);
                const v2f aR = *(const v2f*)(aT + 1 * MROWS * KST);
                const v2f aN = *(const v2f*)(aT + 2 * MROWS * KST);
                accC[t] = __builtin_amdgcn_wmma_f32_16x16x4_f32(false, aC, false, bCf, (short)0, accC[t], false, false);
                accR[t] = __builtin_amdgcn_wmma_f32_16x16x4_f32(false, aR, false, bRf, (short)0, accR[t], false, false);
                accN[t] = __builtin_amdgcn_wmma_f32_16x16x4_f32(false, aN, false, bRf, (short)0, accN[t], false, false);
            }
        }
    }

    // ---- fused epilogue: row-wise dot products over D ----
    // C/D layout: VGPR v, lane L -> (M = v + 8*(L/16), N = L%16)
    #pragma unroll
    for (int t = 0; t < 2; ++t) {
        float pos[8], negv[8];
        #pragma unroll
        for (int v = 0; v < 8; ++v) {
            pos[v]  = accC[t][v] * accR[t][v];
            negv[v] = accC[t][v] * accN[t][v];
        }
        #pragma unroll
        for (int m = 1; m < 16; m <<= 1) {
            #pragma unroll
            for (int v = 0; v < 8; ++v) {
                pos[v]  += __shfl_xor(pos[v],  m, 32);
                negv[v] += __shfl_xor(negv[v], m, 32);
            }
        }
        if (l16 == 0) {
            #pragma unroll
            for (int v = 0; v < 8; ++v) {
                const int row = t * 16 + v + (halfId << 3);
                atomicAdd(&ldsPos[row], pos[v]);
                atomicAdd(&ldsNeg[row], negv[v]);
            }
        }
    }
    __syncthreads();
    if (tid < MROWS) {
        out[rowBase + tid]          = ldsPos[tid];   // f_pos
        out[N_ROWS + rowBase + tid] = ldsNeg[tid];   // f_neg
    }
}

// Deterministic hinge-loss reduction over the already-written f_pos/f_neg.
__global__ __launch_bounds__(256) void loss_kernel(float* __restrict__ out)
{
    __shared__ float red[256];
    float local = 0.f;
    for (int r = threadIdx.x; r < N_ROWS; r += 256) {
        const float d = out[N_ROWS + r] - out[r] + MARGIN;
        local += (d > 0.f) ? d : 0.f;
    }
    red[threadIdx.x] = local;
    __syncthreads();
    for (int s = 128; s > 0; s >>= 1) {
        if (threadIdx.x < s) red[threadIdx.x] += red[threadIdx.x + s];
        __syncthreads();
    }
    if (threadIdx.x == 0) out[2 * N_ROWS] = red[0];
}

extern "C" void kernel_launch(void* const* d_in, const int* in_sizes, int n_in,
                              void* d_out, int out_size, void* d_ws, size_t ws_size,
                              hipStream_t stream)
{
    const float* ctx  = (const float*)d_in[0];
    const float* rsp  = (const float*)d_in[1];
    const float* ngr  = (const float*)d_in[2];
    const float* embC = (const float*)d_in[3];
    const float* embR = (const float*)d_in[4];
    float* out = (float*)d_out;

    embed_gemm_kernel<<<N_ROWS / MROWS, 256, 0, stream>>>(ctx, rsp, ngr, embC, embR, out);
    loss_kernel<<<1, 256, 0, stream>>>(out);
}